// ProbToLabel_37873021616310
// MI455X (gfx1250) — compile-verified
//
#include <hip/hip_runtime.h>

// ProbToLabel: y[i] = label_table[argmax_j inputs[i, j]], C == 8.
//
// Memory-bound streaming kernel: 512 MiB read + 64 MiB write, once each
// -> ~26 us floor at 23.3 TB/s HBM. No matrix op exists in this workload,
// so no WMMA by design; the CDNA5-relevant path is the memory system:
//   * 128-bit non-temporal loads/stores (read-once / write-once streams
//     should not rinse through the 192 MB L2 as regular-temporal)
//   * 4 rows per thread: the ~30-instruction per-thread prologue
//     (ttmp-based workgroup id math, kernarg + table s_loads, bounds check)
//     is amortized over 144 bytes instead of 36, halving dynamic
//     instructions per byte so instruction issue can't shadow the HBM bound
//   * fused argmax+label tracking: strict '>' scan in index order gives
//     jnp.argmax first-occurrence semantics and carries the label directly
//     (table held in SGPRs via s_load_b256), so no per-row gather exists.

typedef __attribute__((ext_vector_type(4))) float v4f;
typedef __attribute__((ext_vector_type(4))) int   v4i;

__device__ __forceinline__ int argmax_label8(v4f a, v4f b,
                                             int l0, int l1, int l2, int l3,
                                             int l4, int l5, int l6, int l7)
{
    float best = a.x;
    int   lab  = l0;
    { bool c = a.y > best; best = c ? a.y : best; lab = c ? l1 : lab; }
    { bool c = a.z > best; best = c ? a.z : best; lab = c ? l2 : lab; }
    { bool c = a.w > best; best = c ? a.w : best; lab = c ? l3 : lab; }
    { bool c = b.x > best; best = c ? b.x : best; lab = c ? l4 : lab; }
    { bool c = b.y > best; best = c ? b.y : best; lab = c ? l5 : lab; }
    { bool c = b.z > best; best = c ? b.z : best; lab = c ? l6 : lab; }
    { bool c = b.w > best;                        lab = c ? l7 : lab; }
    return lab;
}

__global__ __launch_bounds__(256) void ProbToLabel_kernel(
    const float* __restrict__ in,      // [n_rows, 8] f32
    const int*   __restrict__ table,   // [8] i32 (uniform -> s_load_b256)
    int*         __restrict__ out,     // [n_rows] i32
    int n_rows)
{
    const int grp  = blockIdx.x * 256 + threadIdx.x;  // group of 4 rows
    const int row0 = grp * 4;
    if (row0 >= n_rows) return;

    // Wave-uniform label table -> one s_load_b256; selects use SGPR operands.
    const int l0 = table[0], l1 = table[1], l2 = table[2], l3 = table[3];
    const int l4 = table[4], l5 = table[5], l6 = table[6], l7 = table[7];

    const v4f* p = (const v4f*)(in + (size_t)row0 * 8);

    if (row0 + 3 < n_rows) {
        // Fast path: 8 clause-grouped b128 NT loads cover 128B/thread;
        // the wave's clause streams a contiguous 4KB block.
        v4f a0 = __builtin_nontemporal_load(p + 0);
        v4f b0 = __builtin_nontemporal_load(p + 1);
        v4f a1 = __builtin_nontemporal_load(p + 2);
        v4f b1 = __builtin_nontemporal_load(p + 3);
        v4f a2 = __builtin_nontemporal_load(p + 4);
        v4f b2 = __builtin_nontemporal_load(p + 5);
        v4f a3 = __builtin_nontemporal_load(p + 6);
        v4f b3 = __builtin_nontemporal_load(p + 7);

        v4i res;
        res.x = argmax_label8(a0, b0, l0, l1, l2, l3, l4, l5, l6, l7);
        res.y = argmax_label8(a1, b1, l0, l1, l2, l3, l4, l5, l6, l7);
        res.z = argmax_label8(a2, b2, l0, l1, l2, l3, l4, l5, l6, l7);
        res.w = argmax_label8(a3, b3, l0, l1, l2, l3, l4, l5, l6, l7);

        // One 128-bit NT store per thread (512B contiguous per wave).
        __builtin_nontemporal_store(res, (v4i*)(out + row0));
    } else {
        // Tail (never taken for N = 2^24; kept for generality).
        for (int r = row0; r < n_rows; ++r) {
            const v4f* q = (const v4f*)(in + (size_t)r * 8);
            v4f a = __builtin_nontemporal_load(q);
            v4f b = __builtin_nontemporal_load(q + 1);
            int lab = argmax_label8(a, b, l0, l1, l2, l3, l4, l5, l6, l7);
            __builtin_nontemporal_store(lab, out + r);
        }
    }
}

extern "C" void kernel_launch(void* const* d_in, const int* in_sizes, int n_in,
                              void* d_out, int out_size, void* d_ws, size_t ws_size,
                              hipStream_t stream) {
    const float* in    = (const float*)d_in[0];   // [N, 8] f32
    const int*   table = (const int*)d_in[1];     // [8] i32
    int*         out   = (int*)d_out;             // [N] i32 output

    const int n_rows = out_size;                  // N = 16,777,216
    const int block  = 256;                       // 8 wave32s per block
    const int groups = (n_rows + 3) / 4;          // 4 rows per thread
    const int grid   = (groups + block - 1) / block;

    ProbToLabel_kernel<<<grid, block, 0, stream>>>(in, table, out, n_rows);
}